// SOPACell_88923002896974
// MI455X (gfx1250) — compile-verified
//
#include <hip/hip_runtime.h>
#include <hip/hip_bf16.h>

// SOPA cell, MI455X (gfx1250, wave32).
// Pass 1: fp32 WMMA GEMM  u5[t][b][dir][k][n] = x @ W  (only the 5 used gates)
// Pass 2: per-channel sequential scan (8192 channels), dir=1 time-reversed.

#define L_T   2048
#define B_B   16
#define N_IN  256
#define N_OUT 256
#define KG    6      // gates in weight matrix
#define KU    5      // gates actually used
#define NCOL  (2 * KU * N_OUT)      // 2560: GEMM N dimension (dir,k,n)

typedef float v2f __attribute__((ext_vector_type(2)));
typedef float v8f __attribute__((ext_vector_type(8)));

// ---------------------------------------------------------------------------
// Kernel 1: u5 = x @ W, fp32 WMMA 16x16x4.
// Grid: one block per timestep t (2048 blocks).  Block: 256 threads = 8 waves.
// M-tile for block t is rows (t*16 + b), b = 0..15  (exactly one 16-row tile).
// 160 N-tiles = (dir:2) x (k:5) x (n-tile:16).  Each wave handles 20 of them.
// ---------------------------------------------------------------------------
__global__ __launch_bounds__(256)
void sopa_gemm_f32wmma(const float* __restrict__ x,
                       const float* __restrict__ w,   // [256, 3072]
                       float* __restrict__ u5)        // [L, B, 2, 5, 256]
{
    // x tile staged in LDS; row pitch 260 (=256+4) so the 16 lanes of an A
    // fragment hit distinct banks (4*m mod 64 are all distinct for m=0..15).
    __shared__ float xs[16][260];

    const int t    = blockIdx.x;
    const int tid  = threadIdx.x;

    for (int i = tid; i < 16 * 256; i += 256) {
        const int r = i >> 8;
        const int c = i & 255;
        xs[r][c] = x[((size_t)t * 16 + r) * N_IN + c];
    }
    __syncthreads();

    const int wave = tid >> 5;
    const int lane = tid & 31;
    const int lm   = lane & 15;   // M-row (A) / N-col (B,C) within the tile
    const int lh   = lane >> 4;   // half-wave select (K pairs / M+8)

    for (int jt = wave; jt < 160; jt += 8) {
        const int dir = jt / 80;
        const int rem = jt % 80;
        const int k   = rem >> 4;          // gate 0..4
        const int n0  = (rem & 15) << 4;   // n-tile base

        // Weight column for this lane's N position: j = dir*1536 + n*6 + k
        const float* wcol = w + (size_t)dir * (N_OUT * KG) + (size_t)(n0 + lm) * KG + k;

        v8f acc = {0.f, 0.f, 0.f, 0.f, 0.f, 0.f, 0.f, 0.f};

        #pragma unroll 4
        for (int k0 = 0; k0 < N_IN; k0 += 4) {
            const int ka = k0 + lh * 2;
            // A fragment (16x4): lane lm holds row lm, K = {ka, ka+1}
            v2f a;
            a.x = xs[lm][ka];
            a.y = xs[lm][ka + 1];
            // B fragment (4x16): lane holds col (n0+lm), K rows {ka, ka+1}
            v2f bb;
            bb.x = wcol[(size_t)ka * (2 * N_OUT * KG)];
            bb.y = wcol[(size_t)(ka + 1) * (2 * N_OUT * KG)];
            acc = __builtin_amdgcn_wmma_f32_16x16x4_f32(
                false, a, false, bb, (short)0, acc, false, false);
        }

        // C/D layout: VGPR v -> M = v + lh*8 (== batch index b), N = lm.
        const size_t jp   = (size_t)(dir * KU + k) * N_OUT + n0 + lm;
        const size_t base = (size_t)t * 16 * NCOL + jp;
        #pragma unroll
        for (int v = 0; v < 8; ++v) {
            const int b = v + lh * 8;
            u5[base + (size_t)b * NCOL] = acc[v];
        }
    }
}

// ---------------------------------------------------------------------------
// Kernel 2: sequential recurrence, one thread per (b, dir, n) channel.
//   c1' = (c1 - x1)*f1 + x1
//   c2' = (c2 - x3*d)*f2 + x3*d ;  d' = x2
// dir==1 walks t from L-1 down to 0 (equivalent to flip/scan/flip).
// ---------------------------------------------------------------------------
__global__ __launch_bounds__(256)
void sopa_scan(const float* __restrict__ u5,
               const float* __restrict__ bias,   // [6, 2, 256]
               const float* __restrict__ c1i,    // [16, 512]
               const float* __restrict__ c2i,
               const float* __restrict__ di,
               float* __restrict__ out)
{
    const int ch  = blockIdx.x * blockDim.x + threadIdx.x;  // 0..8191
    const int n   = ch & 255;
    const int dir = (ch >> 8) & 1;
    const int b   = ch >> 9;

    const int dn = dir * 256 + n;
    const float b3 = bias[3 * 512 + dn];
    const float b4 = bias[4 * 512 + dn];

    float c1 = c1i[b * 512 + dn];
    float c2 = c2i[b * 512 + dn];
    float d  = di [b * 512 + dn];

    const size_t colbase = (size_t)(dir * KU) * N_OUT + n;   // + k*256
    const size_t outc    = ((size_t)b * 2 + dir) * 256 + n;  // within one t
    const size_t LB512   = (size_t)L_T * 16 * 512;           // c1s block size

    int t = dir ? (L_T - 1) : 0;
    const int dt = dir ? -1 : 1;

    for (int s = 0; s < L_T; ++s, t += dt) {
        const float* up = u5 + ((size_t)t * 16 + b) * NCOL + colbase;
        // prefetch a few steps ahead on the sequential walk
        __builtin_prefetch(up + (ptrdiff_t)dt * 4 * 16 * NCOL, 0, 0);

        const float x1 = up[0];
        const float x2 = up[256];
        const float x3 = up[512];
        const float f1 = 1.0f / (1.0f + __expf(-(up[768]  + b3)));
        const float f2 = 1.0f / (1.0f + __expf(-(up[1024] + b4)));

        c1 = (c1 - x1) * f1 + x1;
        const float tmp = x3 * d;
        c2 = (c2 - tmp) * f2 + tmp;
        d  = x2;

        const size_t o = (size_t)t * (16 * 512) + outc;
        out[o]         = c1;   // c1s
        out[LB512 + o] = c2;   // c2s
    }

    // finals: c1_f, c2_f, d_f each [16, 512]
    const size_t fin = 2 * LB512;
    out[fin +          (size_t)b * 512 + dn] = c1;
    out[fin +  8192 +  (size_t)b * 512 + dn] = c2;
    out[fin + 16384 +  (size_t)b * 512 + dn] = d;
}

// ---------------------------------------------------------------------------
extern "C" void kernel_launch(void* const* d_in, const int* in_sizes, int n_in,
                              void* d_out, int out_size, void* d_ws, size_t ws_size,
                              hipStream_t stream)
{
    const float* x    = (const float*)d_in[0];  // [2048, 16, 256]
    const float* w    = (const float*)d_in[1];  // [256, 3072]
    const float* bias = (const float*)d_in[2];  // [3072]
    const float* c1i  = (const float*)d_in[3];  // [16, 512]
    const float* c2i  = (const float*)d_in[4];
    const float* di   = (const float*)d_in[5];
    float* out = (float*)d_out;

    // workspace: u5 [32768, 2560] fp32 = 335,544,320 bytes
    float* u5 = (float*)d_ws;

    sopa_gemm_f32wmma<<<L_T, 256, 0, stream>>>(x, w, u5);
    sopa_scan<<<(B_B * 2 * N_OUT) / 256, 256, 0, stream>>>(u5, bias, c1i, c2i, di, out);
}